// DCNv4_Block_9938554323222
// MI455X (gfx1250) — compile-verified
//
#include <hip/hip_runtime.h>
#include <math.h>

typedef float v2f __attribute__((ext_vector_type(2)));
typedef float v8f __attribute__((ext_vector_type(8)));
typedef int   v4i __attribute__((vector_size(4 * sizeof(int))));

#define Ndim   4
#define Cdim   256
#define Hdim   64
#define Wdim   64
#define HW     (Hdim * Wdim)        // 4096
#define Mrows  (Ndim * HW)          // 16384
#define Gdim   16
#define Kk     3
#define K2v    9
#define Cg     (Cdim / Gdim)        // 16
#define NOM    (Gdim * K2v * 3)     // 432
#define EPSV   1e-5f

#if defined(__AMDGCN__) && __has_builtin(__builtin_amdgcn_global_load_async_to_lds_b128)
#define HAVE_ASYNC_LDS 1
#else
#define HAVE_ASYNC_LDS 0
#endif

__device__ __forceinline__ v8f wmma_f32x4(v2f a, v2f b, v8f c) {
  // D = A(16x4,f32) * B(4x16,f32) + C(16x16,f32) -> v_wmma_f32_16x16x4_f32
  return __builtin_amdgcn_wmma_f32_16x16x4_f32(false, a, false, b, (short)0, c,
                                               false, false);
}

__device__ __forceinline__ void async_copy16(const float* src, float* lds_dst) {
#if HAVE_ASYNC_LDS
  __builtin_amdgcn_global_load_async_to_lds_b128(
      (v4i*)const_cast<float*>(src), (v4i*)lds_dst, 0, 0);
#else
  const float4 d = *(const float4*)src;
  *(float4*)lds_dst = d;
#endif
}

__device__ __forceinline__ void async_join() {
#if HAVE_ASYNC_LDS
  asm volatile("s_wait_asynccnt 0x0" ::: "memory");
#endif
}

// ---------------------------------------------------------------------------
// WMMA tile worker for the transposed-A panel (As_t layout: [k=0..255][m=0..31]
// padded to stride 48 -> lanes 0-15 / 16-31 hit disjoint LDS bank sets).
// One call: 2 accumulators (M rows m0..m0+15 and m0+16..m0+31) x 16 N cols.
// LDB is compile-time so all B addresses are immediate offsets (no v_mul_u64).
// ---------------------------------------------------------------------------
template <int LDB>
__device__ __forceinline__ void gemm_tiles_xT(const float* __restrict__ As_t,
                                              const float* __restrict__ B,
                                              const float* __restrict__ bias,
                                              float* __restrict__ Cout,
                                              int m0, int col0, int lane)
{
  const int lo  = lane & 15;
  const int hiK = (lane >> 4) << 1;            // 0 / 2
  v8f acc0 = {0.f, 0.f, 0.f, 0.f, 0.f, 0.f, 0.f, 0.f};
  v8f acc1 = acc0;
  const float* bp = B + (size_t)hiK * LDB + col0 + lo;
  const float* ap = As_t + hiK * 48;

  for (int k4 = 0; k4 < Cdim; k4 += 16) {
    v2f b[4], a0[4], a1[4];
    #pragma unroll
    for (int j = 0; j < 4; ++j) {              // batch 12 loads ahead of WMMAs
      const int kk = k4 + 4 * j;
      b[j].x  = bp[(size_t)kk * LDB];
      b[j].y  = bp[(size_t)(kk + 1) * LDB];
      a0[j].x = ap[kk * 48 + lo];
      a0[j].y = ap[(kk + 1) * 48 + lo];
      a1[j].x = ap[kk * 48 + 16 + lo];
      a1[j].y = ap[(kk + 1) * 48 + 16 + lo];
    }
    #pragma unroll
    for (int j = 0; j < 4; ++j) {
      acc0 = wmma_f32x4(a0[j], b[j], acc0);
      acc1 = wmma_f32x4(a1[j], b[j], acc1);
    }
  }

  const float bval = bias[col0 + lo];
  const int mB = m0 + ((lane >> 4) << 3);      // +8 rows for lanes 16-31
  #pragma unroll
  for (int r = 0; r < 8; ++r) {
    Cout[(size_t)(mB + r) * LDB + col0 + lo]      = acc0[r] + bval;
    Cout[(size_t)(mB + 16 + r) * LDB + col0 + lo] = acc1[r] + bval;
  }
}

// ---------------------------------------------------------------------------
// Fused projection GEMM: [16384 x 256](NCHW x) @ [Wv | Wom] + [bv | bom]
// Block = 32 M-rows; A panel staged transposed via async b128 copies.
// ---------------------------------------------------------------------------
__global__ __launch_bounds__(256)
void gemm_x_fused(const float* __restrict__ x,  const float* __restrict__ Wv,
                  const float* __restrict__ bv, const float* __restrict__ Wom,
                  const float* __restrict__ bom, float* __restrict__ v,
                  float* __restrict__ om)
{
  __shared__ float As_t[Cdim * 48];            // [k][m] padded, 48 KB
  const int m0    = blockIdx.x * 32;
  const int n_img = m0 / HW;                   // 32-row panel never crosses n
  const int hw0   = m0 % HW;
  const int t     = threadIdx.x;

  // x[(n*256+c)*4096 + hw] is contiguous along hw == contiguous along m:
  // copy 16B runs straight into the transposed panel.
  #pragma unroll
  for (int i = 0; i < 8; ++i) {
    const int j = t + 256 * i;                 // 2048 float4 transfers
    const int c = j >> 3;
    const int q = j & 7;                       // which 16B run of the 32 floats
    const float* src = x + ((size_t)n_img * Cdim + c) * HW + hw0 + q * 4;
    async_copy16(src, &As_t[c * 48 + q * 4]);
  }
  async_join();
  __syncthreads();

  const int wave = t >> 5;
  const int lane = t & 31;
  for (int tile = wave; tile < 43; tile += 8) {
    if (tile < 16)
      gemm_tiles_xT<Cdim>(As_t, Wv, bv, v, m0, tile * 16, lane);
    else
      gemm_tiles_xT<NOM>(As_t, Wom, bom, om, m0, (tile - 16) * 16, lane);
  }
}

// ---------------------------------------------------------------------------
// Row-major-A WMMA GEMM: o[16384 x 256] = sampled @ Wout + bout
// ---------------------------------------------------------------------------
__global__ __launch_bounds__(256)
void gemm_row(const float* __restrict__ A, const float* __restrict__ B,
              const float* __restrict__ bias, float* __restrict__ C)
{
  __shared__ float As[32 * 260];               // [m][k] padded
  const int m0 = blockIdx.x * 32;
  const int t  = threadIdx.x;

  #pragma unroll
  for (int i = 0; i < 8; ++i) {
    const int j  = t + 256 * i;                // 2048 float4 transfers
    const int r  = j >> 6;
    const int c4 = j & 63;
    async_copy16(A + (size_t)(m0 + r) * Cdim + c4 * 4, &As[r * 260 + c4 * 4]);
  }
  async_join();
  __syncthreads();

  const int wave = t >> 5;
  const int lane = t & 31;
  const int lo   = lane & 15;
  const int hiK  = (lane >> 4) << 1;

  for (int tile = wave; tile < 16; tile += 8) {
    const int n0 = tile * 16;
    v8f acc0 = {0.f, 0.f, 0.f, 0.f, 0.f, 0.f, 0.f, 0.f};
    v8f acc1 = acc0;
    const float* bp  = B + (size_t)hiK * Cdim + n0 + lo;
    const float* ap0 = As + lo * 260 + hiK;
    const float* ap1 = As + (16 + lo) * 260 + hiK;

    for (int k4 = 0; k4 < Cdim; k4 += 16) {
      v2f b[4], a0[4], a1[4];
      #pragma unroll
      for (int j = 0; j < 4; ++j) {
        const int kk = k4 + 4 * j;
        b[j].x = bp[(size_t)kk * Cdim];
        b[j].y = bp[(size_t)(kk + 1) * Cdim];
        a0[j] = *(const v2f*)(ap0 + kk);       // ds_load_b64, conflict-free
        a1[j] = *(const v2f*)(ap1 + kk);
      }
      #pragma unroll
      for (int j = 0; j < 4; ++j) {
        acc0 = wmma_f32x4(a0[j], b[j], acc0);
        acc1 = wmma_f32x4(a1[j], b[j], acc1);
      }
    }

    const float bval = bias[n0 + lo];
    const int mB = m0 + ((lane >> 4) << 3);
    #pragma unroll
    for (int r = 0; r < 8; ++r) {
      C[(size_t)(mB + r) * Cdim + n0 + lo]      = acc0[r] + bval;
      C[(size_t)(mB + 16 + r) * Cdim + n0 + lo] = acc1[r] + bval;
    }
  }
}

// ---------------------------------------------------------------------------
// DCNv4 deformable bilinear sampling. One thread = one (n,h,w,g); Cg=16 ch.
// ---------------------------------------------------------------------------
__global__ __launch_bounds__(256)
void dcn_sample(const float* __restrict__ v, const float* __restrict__ om,
                float* __restrict__ out)
{
  const int idx = blockIdx.x * 256 + threadIdx.x;   // N*HW*G
  const int g   = idx & 15;
  const int nhw = idx >> 4;
  const int n   = nhw >> 12;
  const int hw  = nhw & 4095;
  const int h   = hw >> 6;
  const int w   = hw & 63;

  const float* op    = om + (size_t)nhw * NOM + g * (K2v * 3);
  const float* vbase = v + (size_t)n * HW * Cdim + g * Cg;

  float acc[Cg];
  #pragma unroll
  for (int c = 0; c < Cg; ++c) acc[c] = 0.f;

  auto corner = [&](int yy, int xx, float wt) {
    if (yy >= 0 && yy < Hdim && xx >= 0 && xx < Wdim) {
      const float4* p = (const float4*)(vbase + (size_t)(yy * Wdim + xx) * Cdim);
      #pragma unroll
      for (int q = 0; q < 4; ++q) {
        const float4 d = p[q];
        acc[4 * q + 0] += wt * d.x;
        acc[4 * q + 1] += wt * d.y;
        acc[4 * q + 2] += wt * d.z;
        acc[4 * q + 3] += wt * d.w;
      }
    }
  };

  #pragma unroll
  for (int k = 0; k < K2v; ++k) {
    const float offy = op[2 * k];
    const float offx = op[2 * k + 1];
    const float msk  = op[2 * K2v + k];
    const float sy = (float)(h + (k / Kk) - 1) + offy;
    const float sx = (float)(w + (k % Kk) - 1) + offx;
    const float y0f = floorf(sy), x0f = floorf(sx);
    const float fy = sy - y0f, fx = sx - x0f;
    const int y0 = (int)y0f, x0 = (int)x0f;
    corner(y0,     x0,     (1.f - fy) * (1.f - fx) * msk);
    corner(y0,     x0 + 1, (1.f - fy) * fx * msk);
    corner(y0 + 1, x0,     fy * (1.f - fx) * msk);
    corner(y0 + 1, x0 + 1, fy * fx * msk);
  }

  float* o = out + (size_t)nhw * Cdim + g * Cg;
  #pragma unroll
  for (int q = 0; q < 4; ++q)
    ((float4*)o)[q] =
        make_float4(acc[4 * q], acc[4 * q + 1], acc[4 * q + 2], acc[4 * q + 3]);
}

// ---------------------------------------------------------------------------
// Deterministic two-stage per-channel mean/var reduction (no float atomics).
// ---------------------------------------------------------------------------
__global__ __launch_bounds__(256)
void stats_partial(const float* __restrict__ o, float* __restrict__ part)
{
  const int ch = threadIdx.x;
  const int b  = blockIdx.x;                 // 256 blocks x 64 rows
  float s = 0.f, s2 = 0.f;
  for (int r = b * 64; r < (b + 1) * 64; ++r) {
    const float val = o[(size_t)r * Cdim + ch];
    s  += val;
    s2 += val * val;
  }
  part[b * Cdim + ch]              = s;
  part[256 * Cdim + b * Cdim + ch] = s2;
}

__global__ __launch_bounds__(256)
void stats_reduce(const float* __restrict__ part, float* __restrict__ stats)
{
  const int ch = threadIdx.x;
  float s = 0.f, s2 = 0.f;
  for (int b = 0; b < 256; ++b) {
    s  += part[b * Cdim + ch];
    s2 += part[256 * Cdim + b * Cdim + ch];
  }
  const float inv_n = 1.f / (float)Mrows;
  const float mean  = s * inv_n;
  const float var   = s2 * inv_n - mean * mean;
  stats[ch]        = mean;
  stats[Cdim + ch] = rsqrtf(var + EPSV);
}

// ---------------------------------------------------------------------------
// Normalize + SiLU + NHWC -> NCHW transpose (writes coalesced in w).
// ---------------------------------------------------------------------------
__global__ __launch_bounds__(256)
void finalize(const float* __restrict__ o, const float* __restrict__ stats,
              const float* __restrict__ gamma, const float* __restrict__ beta,
              float* __restrict__ out)
{
  const int idx = blockIdx.x * 256 + threadIdx.x;   // N*C*H*W
  const int w   = idx & 63;
  const int h   = (idx >> 6) & 63;
  const int ch  = (idx >> 12) & 255;
  const int n   = idx >> 20;
  const float val = o[((size_t)n * HW + h * Wdim + w) * Cdim + ch];
  float y = gamma[ch] * (val - stats[ch]) * stats[Cdim + ch] + beta[ch];
  y = y * (1.f / (1.f + __expf(-y)));
  out[idx] = y;
}

// ---------------------------------------------------------------------------
extern "C" void kernel_launch(void* const* d_in, const int* in_sizes, int n_in,
                              void* d_out, int out_size, void* d_ws,
                              size_t ws_size, hipStream_t stream)
{
  const float* x     = (const float*)d_in[0];
  const float* Wv    = (const float*)d_in[1];
  const float* bv    = (const float*)d_in[2];
  const float* Wom   = (const float*)d_in[3];
  const float* bom   = (const float*)d_in[4];
  const float* Wout  = (const float*)d_in[5];
  const float* bout  = (const float*)d_in[6];
  const float* gamma = (const float*)d_in[7];
  const float* beta  = (const float*)d_in[8];

  float* ws    = (float*)d_ws;
  float* v     = ws;                                  // 16384*256
  float* om    = v   + (size_t)Mrows * Cdim;          // 16384*432
  float* smp   = om  + (size_t)Mrows * NOM;           // 16384*256
  float* o     = smp + (size_t)Mrows * Cdim;          // 16384*256
  float* part  = o   + (size_t)Mrows * Cdim;          // 2*256*256
  float* stats = part + 2 * 256 * Cdim;               // 512

  gemm_x_fused <<<Mrows / 32, 256, 0, stream>>>(x, Wv, bv, Wom, bom, v, om);
  dcn_sample   <<<(Mrows * Gdim) / 256, 256, 0, stream>>>(v, om, smp);
  gemm_row     <<<Mrows / 32, 256, 0, stream>>>(smp, Wout, bout, o);
  stats_partial<<<256, 256, 0, stream>>>(o, part);
  stats_reduce <<<1, 256, 0, stream>>>(part, stats);
  finalize     <<<(Mrows * Cdim) / 256, 256, 0, stream>>>(o, stats, gamma, beta,
                                                          (float*)d_out);
}